// SimpleTP_70446053589162
// MI455X (gfx1250) — compile-verified
//
#include <hip/hip_runtime.h>

typedef float v2f __attribute__((ext_vector_type(2)));
typedef float v8f __attribute__((ext_vector_type(8)));

#define NBLK 1024
#define TPB  256
#define PSTRIDE 16   // padded partials row: cols 0..5 = data, 6..15 = 0.0f

// ---------------------------------------------------------------------------
// Kernel 1: streaming Gram accumulation. Each thread owns contiguous 48-byte
// groups (4 points = 3 x float4), accumulates the 6 unique entries of
// S = sum_n x_n x_n^T in f32 registers, then wave32 butterfly + LDS reduce.
// Writes a 16-float zero-padded partials row per block (so the WMMA finalize
// kernel can load B unconditionally with no EXEC masking).
// ---------------------------------------------------------------------------
__global__ __launch_bounds__(TPB) void gram_partial_kernel(
    const float* __restrict__ frac, float* __restrict__ partials, int n_points) {
  const int tid      = blockIdx.x * blockDim.x + threadIdx.x;
  const int nthreads = gridDim.x * blockDim.x;
  const int n_groups = n_points >> 2;

  float s00 = 0.f, s11 = 0.f, s22 = 0.f, s01 = 0.f, s02 = 0.f, s12 = 0.f;

  const float4* __restrict__ f4 = reinterpret_cast<const float4*>(frac);
  for (int g = tid; g < n_groups; g += nthreads) {
    // Speculative prefetch of next tile (dropped silently if OOB).
    __builtin_prefetch((const void*)(f4 + 3 * (size_t)(g + nthreads)), 0, 0);
    const float4 a = f4[3 * g + 0];
    const float4 b = f4[3 * g + 1];
    const float4 c = f4[3 * g + 2];
    // points: (a.x,a.y,a.z) (a.w,b.x,b.y) (b.z,b.w,c.x) (c.y,c.z,c.w)
    s00 = fmaf(a.x, a.x, s00); s11 = fmaf(a.y, a.y, s11); s22 = fmaf(a.z, a.z, s22);
    s01 = fmaf(a.x, a.y, s01); s02 = fmaf(a.x, a.z, s02); s12 = fmaf(a.y, a.z, s12);

    s00 = fmaf(a.w, a.w, s00); s11 = fmaf(b.x, b.x, s11); s22 = fmaf(b.y, b.y, s22);
    s01 = fmaf(a.w, b.x, s01); s02 = fmaf(a.w, b.y, s02); s12 = fmaf(b.x, b.y, s12);

    s00 = fmaf(b.z, b.z, s00); s11 = fmaf(b.w, b.w, s11); s22 = fmaf(c.x, c.x, s22);
    s01 = fmaf(b.z, b.w, s01); s02 = fmaf(b.z, c.x, s02); s12 = fmaf(b.w, c.x, s12);

    s00 = fmaf(c.y, c.y, s00); s11 = fmaf(c.z, c.z, s11); s22 = fmaf(c.w, c.w, s22);
    s01 = fmaf(c.y, c.z, s01); s02 = fmaf(c.y, c.w, s02); s12 = fmaf(c.z, c.w, s12);
  }

  // Tail points (n_points % 4), handled once by thread 0.
  if (tid == 0) {
    for (int p = n_groups << 2; p < n_points; ++p) {
      const float x = frac[3 * p + 0];
      const float y = frac[3 * p + 1];
      const float z = frac[3 * p + 2];
      s00 = fmaf(x, x, s00); s11 = fmaf(y, y, s11); s22 = fmaf(z, z, s22);
      s01 = fmaf(x, y, s01); s02 = fmaf(x, z, s02); s12 = fmaf(y, z, s12);
    }
  }

  // wave32 butterfly reduction (fixed order -> deterministic)
  #pragma unroll
  for (int off = 16; off > 0; off >>= 1) {
    s00 += __shfl_xor(s00, off, 32);
    s11 += __shfl_xor(s11, off, 32);
    s22 += __shfl_xor(s22, off, 32);
    s01 += __shfl_xor(s01, off, 32);
    s02 += __shfl_xor(s02, off, 32);
    s12 += __shfl_xor(s12, off, 32);
  }

  __shared__ float red[TPB / 32][6];
  const int wave = threadIdx.x >> 5;
  const int lane = threadIdx.x & 31;
  if (lane == 0) {
    red[wave][0] = s00; red[wave][1] = s11; red[wave][2] = s22;
    red[wave][3] = s01; red[wave][4] = s02; red[wave][5] = s12;
  }
  __syncthreads();
  if (threadIdx.x < PSTRIDE) {
    float s = 0.f;
    if (threadIdx.x < 6) {
      #pragma unroll
      for (int w = 0; w < TPB / 32; ++w) s += red[w][threadIdx.x];
    }
    // cols 6..15 written as 0.0f -> finalize kernel loads need no masking
    partials[blockIdx.x * PSTRIDE + threadIdx.x] = s;
  }
}

// ---------------------------------------------------------------------------
// Kernel 2: final reduction as a ones-vector GEMM on the matrix core.
// out_cols = ones[1 x nrows] * P[nrows x 16] via V_WMMA_F32_16X16X4_F32:
//   A = all-ones 16x4 (every lane {1,1}); B = 4 partial rows per WMMA
//   (VGPR0: rows r,r+2 across lane halves; VGPR1: rows r+1,r+3).
// With A==1, D[0][j] = sum over all K rows of column j, independent of the
// K-row permutation, so this layout is exact. Rows are zero-padded to 16
// columns, so every lane's load is valid (no EXEC masking). Four independent
// accumulators (16 rows/iter) hide the XDL RAW latency; they are combined
// with element-wise adds at the end (fixed order -> deterministic).
// Single full wave (32 threads) => EXEC all ones as WMMA requires.
// ---------------------------------------------------------------------------
__global__ __launch_bounds__(32) void gram_finalize_kernel(
    const float* __restrict__ partials, int nrows, float* __restrict__ out) {
  const int lane = threadIdx.x;   // 0..31, one full wave
  const int col  = lane & 15;     // B/D column index
  const int half = lane >> 4;     // 0: lanes 0-15, 1: lanes 16-31

  const v2f onesA = {1.f, 1.f};
  v8f acc0 = {0.f, 0.f, 0.f, 0.f, 0.f, 0.f, 0.f, 0.f};
  v8f acc1 = acc0, acc2 = acc0, acc3 = acc0;

  // Per-lane base address for its column within a row-pair.
  const float* __restrict__ p = partials + (half * 2) * PSTRIDE + col;

  for (int k = 0; k < nrows; k += 16) {
    const int base = k * PSTRIDE;
    {
      const v2f B = {p[base + 0 * PSTRIDE], p[base + 1 * PSTRIDE]};
      acc0 = __builtin_amdgcn_wmma_f32_16x16x4_f32(false, onesA, false, B,
                                                   (short)0, acc0, false, false);
    }
    {
      const v2f B = {p[base + 4 * PSTRIDE], p[base + 5 * PSTRIDE]};
      acc1 = __builtin_amdgcn_wmma_f32_16x16x4_f32(false, onesA, false, B,
                                                   (short)0, acc1, false, false);
    }
    {
      const v2f B = {p[base + 8 * PSTRIDE], p[base + 9 * PSTRIDE]};
      acc2 = __builtin_amdgcn_wmma_f32_16x16x4_f32(false, onesA, false, B,
                                                   (short)0, acc2, false, false);
    }
    {
      const v2f B = {p[base + 12 * PSTRIDE], p[base + 13 * PSTRIDE]};
      acc3 = __builtin_amdgcn_wmma_f32_16x16x4_f32(false, onesA, false, B,
                                                   (short)0, acc3, false, false);
    }
  }

  // Combine the 4 independent accumulators (linear op, fixed order).
  const v8f acc = (acc0 + acc1) + (acc2 + acc3);

  // D VGPR0, lanes 0-15: element (M=0, N=lane) = column sum of column `lane`.
  __shared__ float S[6];
  if (lane < 6) S[lane] = acc[0];
  __syncthreads();

  if (lane == 0) {
    const float S00 = S[0], S11 = S[1], S22 = S[2];
    const float S01 = S[3], S02 = S[4], S12 = S[5];
    const float INV_SQRT3 = 0.57735026918962576f;
    const float SQRT2     = 1.41421356237309515f;
    const float INV_SQRT6 = 0.40824829046386302f;
    const float INV_SQRT2 = 0.70710678118654752f;
    out[0] = (S00 + S11 + S22) * INV_SQRT3;          // 0e: (x.x)/sqrt(3)
    out[1] = 0.f; out[2] = 0.f; out[3] = 0.f;        // 1e: cross(x,x) == 0
    out[4] = SQRT2 * S02;                            // 2e m=-2
    out[5] = SQRT2 * S01;                            // 2e m=-1
    out[6] = (2.f * S11 - S00 - S22) * INV_SQRT6;    // 2e m= 0
    out[7] = SQRT2 * S12;                            // 2e m=+1
    out[8] = (S22 - S00) * INV_SQRT2;                // 2e m=+2
  }
}

// ---------------------------------------------------------------------------
extern "C" void kernel_launch(void* const* d_in, const int* in_sizes, int n_in,
                              void* d_out, int out_size, void* d_ws, size_t ws_size,
                              hipStream_t stream) {
  const float* frac = (const float*)d_in[0];
  const int n_points = in_sizes[0] / 3;
  float* partials = (float*)d_ws;      // NBLK * 16 floats = 64 KB scratch
  float* out      = (float*)d_out;     // 9 floats

  gram_partial_kernel<<<NBLK, TPB, 0, stream>>>(frac, partials, n_points);
  gram_finalize_kernel<<<1, 32, 0, stream>>>(partials, NBLK, out);
}